// SSIMLoss_41815801594039
// MI455X (gfx1250) — compile-verified
//
#include <hip/hip_runtime.h>
#include <hip/hip_bf16.h>
#include <stdint.h>

typedef __attribute__((ext_vector_type(2))) float v2f;
typedef __attribute__((ext_vector_type(8))) float v8f;

#define HH 512
#define WW 512
#define PLANES 96          // 32 * 3
#define TR 64              // tile rows per block
#define TC 64              // tile cols per block
#define RR 74              // TR + 10 (halo)
#define RCP 76             // region row stride in floats (TC + 10, padded to 76; all cols written)
#define HROW 17            // padded H-buffer row stride (floats)
#define C1F 0.0001f        // 0.01^2
#define C2F 0.0009f        // 0.03^2

#define P_OFF 0
#define T_OFF (RR * RCP)                             // 5624
#define H_OFF (2 * RR * RCP)                         // 11248
#define H_PER_WAVE (26 * HROW)                       // 442
#define LDS_FLOATS (2 * RR * RCP + 8 * H_PER_WAVE)   // 14784 floats = 59136 B

#if defined(__has_builtin)
#if __has_builtin(__builtin_amdgcn_global_load_async_to_lds_b32) && \
    __has_builtin(__builtin_amdgcn_s_wait_asynccnt)
#define USE_ASYNC_LDS 1
#endif
#endif

__global__ __launch_bounds__(256) void ssim_tile_kernel(const float* __restrict__ pred,
                                                        const float* __restrict__ targ,
                                                        float* __restrict__ partial) {
    __shared__ float lds[LDS_FLOATS];
    const int tid = threadIdx.x;
    const int plane = blockIdx.z;
    const int row0 = blockIdx.y * TR;
    const int col0 = blockIdx.x * TC;
    const float* __restrict__ pbase = pred + (size_t)plane * (HH * WW);
    const float* __restrict__ tbase = targ + (size_t)plane * (HH * WW);

    // ---- cooperative zero-padded region load (74 x 76, ALL columns written) ----
    for (int idx = tid; idx < RR * RCP; idx += 256) {
        int r = idx / RCP;
        int c = idx - r * RCP;
        int gr = row0 - 5 + r;
        int gc = col0 - 5 + c;
        bool ok = (gr >= 0) && (gr < HH) && (gc >= 0) && (gc < WW);
#ifdef USE_ASYNC_LDS
        if (ok) {
            int g = gr * WW + gc;
            __builtin_amdgcn_global_load_async_to_lds_b32(
                (int*)(pbase + g), (int*)&lds[P_OFF + idx], 0, 0);
            __builtin_amdgcn_global_load_async_to_lds_b32(
                (int*)(tbase + g), (int*)&lds[T_OFF + idx], 0, 0);
        } else {
            lds[P_OFF + idx] = 0.0f;
            lds[T_OFF + idx] = 0.0f;
        }
#else
        float pv = 0.0f, tv = 0.0f;
        if (ok) {
            int g = gr * WW + gc;
            pv = pbase[g];
            tv = tbase[g];
        }
        lds[P_OFF + idx] = pv;
        lds[T_OFF + idx] = tv;
#endif
    }
#ifdef USE_ASYNC_LDS
    __builtin_amdgcn_s_wait_asynccnt(0);
#endif
    __syncthreads();

    const int lane = tid & 31;
    const int wv   = tid >> 5;
    const int half = lane >> 4;        // 0 or 1 (K sub-chunk selector)
    const int mn   = lane & 15;        // M (A frag) / N (B frag) index

    // Banded box-filter matrix fragments: Band[j][n] = 1 iff n <= j <= n+10.
    // Identical per-lane values serve as horizontal B-frag and vertical A-frag.
    float band[7][2];
#pragma unroll
    for (int k = 0; k < 7; ++k) {
#pragma unroll
        for (int v = 0; v < 2; ++v) {
            int j = 4 * k + v + 2 * half;
            band[k][v] = (j >= mn && j <= mn + 10) ? 1.0f : 0.0f;
        }
    }

    const int hb = H_OFF + wv * H_PER_WAVE;
    const float inv = 1.0f / 121.0f;
    float lsum = 0.0f;

    // 16 sub-tiles (4x4 grid of 16x16); wave wv handles sidx = wv and wv+8.
    for (int sidx = wv; sidx < 16; sidx += 8) {
        const int sr = (sidx >> 2) * 16;
        const int sc = (sidx & 3) * 16;

        // ---- horizontal pass: 5 quantities x (top rows 0..15, bottom rows 16..25) ----
        v8f ct0 = {}, ct1 = {}, ct2 = {}, ct3 = {}, ct4 = {};
        v8f cb0 = {}, cb1 = {}, cb2 = {}, cb3 = {}, cb4 = {};

#pragma unroll
        for (int k = 0; k < 7; ++k) {
            const int coloff = sc + 4 * k + 2 * half;   // even -> 8B-aligned v2f LDS loads
            const int rt = sr + mn;
            int rb = sr + 16 + mn;
            if (rb > RR - 1) rb = RR - 1;   // clamp: keeps garbage finite; band kills it

            v2f pT = *(const v2f*)&lds[P_OFF + rt * RCP + coloff];
            v2f tT = *(const v2f*)&lds[T_OFF + rt * RCP + coloff];
            v2f pB = *(const v2f*)&lds[P_OFF + rb * RCP + coloff];
            v2f tB = *(const v2f*)&lds[T_OFF + rb * RCP + coloff];

            v2f b;
            b.x = band[k][0];
            b.y = band[k][1];

            v2f a;
            a = pT;        ct0 = __builtin_amdgcn_wmma_f32_16x16x4_f32(false, a, false, b, (short)0, ct0, false, false);
            a = tT;        ct1 = __builtin_amdgcn_wmma_f32_16x16x4_f32(false, a, false, b, (short)0, ct1, false, false);
            a = pT * pT;   ct2 = __builtin_amdgcn_wmma_f32_16x16x4_f32(false, a, false, b, (short)0, ct2, false, false);
            a = tT * tT;   ct3 = __builtin_amdgcn_wmma_f32_16x16x4_f32(false, a, false, b, (short)0, ct3, false, false);
            a = pT * tT;   ct4 = __builtin_amdgcn_wmma_f32_16x16x4_f32(false, a, false, b, (short)0, ct4, false, false);

            a = pB;        cb0 = __builtin_amdgcn_wmma_f32_16x16x4_f32(false, a, false, b, (short)0, cb0, false, false);
            a = tB;        cb1 = __builtin_amdgcn_wmma_f32_16x16x4_f32(false, a, false, b, (short)0, cb1, false, false);
            a = pB * pB;   cb2 = __builtin_amdgcn_wmma_f32_16x16x4_f32(false, a, false, b, (short)0, cb2, false, false);
            a = tB * tB;   cb3 = __builtin_amdgcn_wmma_f32_16x16x4_f32(false, a, false, b, (short)0, cb3, false, false);
            a = pB * tB;   cb4 = __builtin_amdgcn_wmma_f32_16x16x4_f32(false, a, false, b, (short)0, cb4, false, false);
        }

        // ---- vertical pass: O = Band(16x26) x H(26x16) via per-wave LDS H buffer ----
        auto vertical = [&](v8f ctq, v8f cbq) -> v8f {
#pragma unroll
            for (int v = 0; v < 8; ++v) {
                int r = v + 8 * half;
                lds[hb + r * HROW + mn] = ctq[v];
                int r2 = r + 16;
                if (r2 < 26) lds[hb + r2 * HROW + mn] = cbq[v];
            }
            // Same-wave LDS ops are in-order: stores visible to following loads.
            v8f acc = {};
#pragma unroll
            for (int k = 0; k < 7; ++k) {
                int i0 = 4 * k + 2 * half;
                float b0 = (i0 < 26)     ? lds[hb + i0 * HROW + mn]       : 0.0f;
                float b1 = (i0 + 1 < 26) ? lds[hb + (i0 + 1) * HROW + mn] : 0.0f;
                v2f a;
                a.x = band[k][0];
                a.y = band[k][1];
                v2f b;
                b.x = b0;
                b.y = b1;
                acc = __builtin_amdgcn_wmma_f32_16x16x4_f32(false, a, false, b, (short)0, acc, false, false);
            }
            return acc;
        };

        v8f o0 = vertical(ct0, cb0);   // sum(p)
        v8f o1 = vertical(ct1, cb1);   // sum(t)
        v8f o2 = vertical(ct2, cb2);   // sum(p*p)
        v8f o3 = vertical(ct3, cb3);   // sum(t*t)
        v8f o4 = vertical(ct4, cb4);   // sum(p*t)

        // ---- per-pixel SSIM + local sum (8 pixels per lane per sub-tile) ----
#pragma unroll
        for (int v = 0; v < 8; ++v) {
            float mu1 = o0[v] * inv, mu2 = o1[v] * inv;
            float x2  = o2[v] * inv, y2  = o3[v] * inv, xy = o4[v] * inv;
            float mu1s = mu1 * mu1, mu2s = mu2 * mu2, mu12 = mu1 * mu2;
            float s1 = x2 - mu1s, s2 = y2 - mu2s, s12 = xy - mu12;
            float num = (2.0f * mu12 + C1F) * (2.0f * s12 + C2F);
            float den = (mu1s + mu2s + C1F) * (s1 + s2 + C2F);
            lsum += num / den;
        }
    }

    // ---- deterministic block reduction (reuse LDS after barrier) ----
    __syncthreads();
    lds[tid] = lsum;
    __syncthreads();
#pragma unroll
    for (int s = 128; s > 0; s >>= 1) {
        if (tid < s) lds[tid] += lds[tid + s];
        __syncthreads();
    }
    if (tid == 0) {
        int bid = (blockIdx.z * gridDim.y + blockIdx.y) * gridDim.x + blockIdx.x;
        partial[bid] = lds[0];
    }
}

__global__ __launch_bounds__(256) void ssim_reduce_kernel(const float* __restrict__ partial,
                                                          float* __restrict__ out, int nb) {
    __shared__ float red[256];
    const int tid = threadIdx.x;
    float s = 0.0f;
    for (int i = tid; i < nb; i += 256) s += partial[i];   // fixed order -> deterministic
    red[tid] = s;
    __syncthreads();
#pragma unroll
    for (int k = 128; k > 0; k >>= 1) {
        if (tid < k) red[tid] += red[tid + k];
        __syncthreads();
    }
    if (tid == 0) {
        float mean = red[0] / (float)(32.0 * 3.0 * 512.0 * 512.0);
        out[0] = 1.0f - mean;
    }
}

extern "C" void kernel_launch(void* const* d_in, const int* in_sizes, int n_in,
                              void* d_out, int out_size, void* d_ws, size_t ws_size,
                              hipStream_t stream) {
    const float* pred = (const float*)d_in[0];
    const float* targ = (const float*)d_in[1];
    float* out = (float*)d_out;
    float* partial = (float*)d_ws;   // 6144 floats

    dim3 grid(WW / TC, HH / TR, PLANES);   // (8, 8, 96) = 6144 blocks
    ssim_tile_kernel<<<grid, 256, 0, stream>>>(pred, targ, partial);

    const int nb = (WW / TC) * (HH / TR) * PLANES;
    ssim_reduce_kernel<<<1, 256, 0, stream>>>(partial, out, nb);
}